// XORNet_36618891165872
// MI455X (gfx1250) — compile-verified
//
#include <hip/hip_runtime.h>
#include <stdint.h>

typedef float v2f __attribute__((ext_vector_type(2)));
typedef float v8f __attribute__((ext_vector_type(8)));

#define BETA    0.9f
#define TSTEPS  20
#define BATCH   1048576u
#define XSTRIDE (BATCH * 8u)   // bytes between consecutive timesteps of x (B,2) f32

// Each wave handles 32 batch elements as two 16-element "slots" (N dim of WMMA).
// Block = 256 threads = 8 waves = 256 elements. Grid = BATCH/256 = 4096 blocks.
__global__ __launch_bounds__(256)
void snn_wmma_scan(const float* __restrict__ xg,
                   const float* __restrict__ W1,   // (4,2) row-major
                   const float* __restrict__ W2,   // (2,4) row-major
                   float* __restrict__ outg) {
  const unsigned tid  = threadIdx.x;
  const unsigned lane = tid & 31u;
  const unsigned wave = tid >> 5;
  const unsigned m    = lane & 15u;     // N column / matrix row index
  const bool     hi   = lane >= 16u;    // upper lane half (K cols 2,3 of A; rows K+2 of B)

  // staging: [2 buffers][8 waves][32 elems][float2] = 4 KB, wave-private regions
  __shared__ float2 sbuf[2][8][32];

  const unsigned elem = blockIdx.x * 256u + wave * 32u + lane;

  // ---- A operands, built once, live in VGPRs for the whole scan ----
  // A1 = W1 (4x2) in rows 0..3 AND replicated into rows 8..11; K cols 2,3 = 0.
  // A layout (16x4 f32): lane m (lanes 0-15) -> (K0,K1); lane 16+m -> (K2,K3).
  v2f a1;
  {
    const unsigned r  = m & 3u;
    const bool valid  = ((m >> 2) & 1u) == 0u;    // m in {0..3, 8..11}
    a1.x = (valid && !hi) ? W1[r * 2 + 0] : 0.0f;
    a1.y = (valid && !hi) ? W1[r * 2 + 1] : 0.0f;
  }
  // A2 = W2 (2x4) in rows 0,1 AND replicated into rows 8,9; all 4 K cols used.
  v2f a2;
  {
    const unsigned r  = m & 1u;
    const bool valid  = (m & 7u) < 2u;            // m in {0,1,8,9}
    a2.x = valid ? W2[r * 4 + (hi ? 2 : 0)] : 0.0f;
    a2.y = valid ? W2[r * 4 + (hi ? 3 : 1)] : 0.0f;
  }

  // workgroup-relative LDS byte addresses for this lane's copy target
  const unsigned lds_cp0 = (unsigned)(uintptr_t)&sbuf[0][wave][lane];
  const unsigned lds_cp1 = (unsigned)(uintptr_t)&sbuf[1][wave][lane];
  const uint64_t xbase   = (uint64_t)(uintptr_t)xg;

  // prologue: stage x[0] (32 lanes x 8B = 256B contiguous per wave)
  asm volatile("global_load_async_to_lds_b64 %0, %1, %2"
               :: "v"(lds_cp0), "v"(elem * 8u), "s"(xbase) : "memory");

  // Persistent state: membranes live directly in the WMMA D accumulators and
  // are updated IN PLACE (leak+reset fma, then D = A*B + D). Unused accumulator
  // rows (A rows are zero there) remain 0 forever -> no per-step zero fills.
  // Replicated A rows keep lanes 16-31 mirroring lanes 0-15 for all state.
  v8f d1a = {}; v8f d1b = {};          // m1, slot A / slot B (elems +0..15 / +16..31)
  v8f d2a = {}; v8f d2b = {};          // m2
  float s1a[4] = {}; float s1b[4] = {};
  float s2a[2] = {}; float s2b[2] = {};

  float2* __restrict__ out2 = (float2*)outg;

#pragma unroll
  for (int t = 0; t < TSTEPS; ++t) {
    // double buffering: kick off x[t+1] first, then wait for x[t] (in-order
    // async returns => "<=1 outstanding" proves the older copy finished).
    if (t + 1 < TSTEPS) {
      const unsigned goff = elem * 8u + (unsigned)(t + 1) * XSTRIDE;
      asm volatile("global_load_async_to_lds_b64 %0, %1, %2"
                   :: "v"(((t + 1) & 1) ? lds_cp1 : lds_cp0), "v"(goff), "s"(xbase)
                   : "memory");
      asm volatile("s_wait_asynccnt 0x1" ::: "memory");
    } else {
      asm volatile("s_wait_asynccnt 0x0" ::: "memory");
    }

    // B1 operand: element (base+m)'s x-pair broadcast to all lanes via LDS.
    // Upper-half rows (K2,K3) are don't-care since A1 cols 2,3 are zero.
    const float2 xa = sbuf[t & 1][wave][m];        // slot A
    const float2 xb = sbuf[t & 1][wave][m + 16];   // slot B
    v2f b1a; b1a.x = xa.x; b1a.y = xa.y;
    v2f b1b; b1b.x = xb.x; b1b.y = xb.y;

    // layer 1: m1' = beta*m1 - s1 (in place) + x@W1^T (WMMA accumulate)
#pragma unroll
    for (int j = 0; j < 4; ++j) {
      d1a[j] = __builtin_fmaf(BETA, d1a[j], -s1a[j]);
      d1b[j] = __builtin_fmaf(BETA, d1b[j], -s1b[j]);
    }
    d1a = __builtin_amdgcn_wmma_f32_16x16x4_f32(false, a1, false, b1a, (short)0, d1a, false, false);
    d1b = __builtin_amdgcn_wmma_f32_16x16x4_f32(false, a1, false, b1b, (short)0, d1b, false, false);
#pragma unroll
    for (int j = 0; j < 4; ++j) {                  // s1 = H(m1 - 1)
      s1a[j] = (d1a[j] > 1.0f) ? 1.0f : 0.0f;
      s1b[j] = (d1b[j] > 1.0f) ? 1.0f : 0.0f;
    }

    // B2 operand (4x16): VGPR0 = rows K0/K2, VGPR1 = rows K1/K3; state is
    // duplicated across lane halves, so this is two cndmasks per slot.
    v2f b2a; b2a.x = hi ? s1a[2] : s1a[0]; b2a.y = hi ? s1a[3] : s1a[1];
    v2f b2b; b2b.x = hi ? s1b[2] : s1b[0]; b2b.y = hi ? s1b[3] : s1b[1];

    // layer 2: m2' = beta*m2 - s2 (in place) + s1@W2^T (WMMA accumulate)
#pragma unroll
    for (int i = 0; i < 2; ++i) {
      d2a[i] = __builtin_fmaf(BETA, d2a[i], -s2a[i]);
      d2b[i] = __builtin_fmaf(BETA, d2b[i], -s2b[i]);
    }
    d2a = __builtin_amdgcn_wmma_f32_16x16x4_f32(false, a2, false, b2a, (short)0, d2a, false, false);
    d2b = __builtin_amdgcn_wmma_f32_16x16x4_f32(false, a2, false, b2b, (short)0, d2b, false, false);
#pragma unroll
    for (int i = 0; i < 2; ++i) {                  // s2 = H(m2 - 1)
      s2a[i] = (d2a[i] > 1.0f) ? 1.0f : 0.0f;
      s2b[i] = (d2b[i] > 1.0f) ? 1.0f : 0.0f;
    }

    // coalesced full-wave store: lane l emits element base+l (slot A lanes 0-15,
    // slot B's duplicated upper half supplies lanes 16-31). 256B/wave/step.
    float2 sv;
    sv.x = hi ? s2b[0] : s2a[0];
    sv.y = hi ? s2b[1] : s2a[1];
    out2[(size_t)t * BATCH + elem] = sv;
  }
}

extern "C" void kernel_launch(void* const* d_in, const int* in_sizes, int n_in,
                              void* d_out, int out_size, void* d_ws, size_t ws_size,
                              hipStream_t stream) {
  const float* x  = (const float*)d_in[0];   // (T,B,2) f32
  const float* W1 = (const float*)d_in[1];   // (4,2)   f32
  const float* W2 = (const float*)d_in[2];   // (2,4)   f32
  float* out = (float*)d_out;                // (T,B,2) f32

  dim3 grid(BATCH / 256u);
  dim3 block(256);
  hipLaunchKernelGGL(snn_wmma_scan, grid, block, 0, stream, x, W1, W2, out);
}